// DynamicReductionNetworkJittable_17136919511681
// MI455X (gfx1250) — compile-verified
//
#include <hip/hip_runtime.h>
#include <cstdint>
#include <math.h>

#define NN   8192
#define GN   2048
#define KNN  16
#define HID  64
#define F2H  128
#define F3H  96
#define BIGF 1e30f
#define TOPINIT 3.0e38f

typedef __attribute__((ext_vector_type(2))) float v2f;
typedef __attribute__((ext_vector_type(8))) float v8f;

__device__ __forceinline__ float eluf(float x){ return x > 0.f ? x : (expf(x) - 1.f); }

__device__ __forceinline__ v8f wmma4(v2f a, v2f b, v8f c){
    // D = A(16x4,f32) * B(4x16,f32) + C(16x16,f32)  -> v_wmma_f32_16x16x4_f32
    return __builtin_amdgcn_wmma_f32_16x16x4_f32(false, a, false, b, (short)0, c, false, false);
}

// ---------------------------------------------------------------- utilities
__global__ void k_init_valid(int* valid){
    int i = blockIdx.x * blockDim.x + threadIdx.x;
    if (i < NN) valid[i] = 1;
}

__global__ void k_clear(unsigned* adj, float* xn){
    int i = blockIdx.x * blockDim.x + threadIdx.x;
    if (i < NN * 64) { adj[i] = 0u; xn[i] = 0.f; }   // NN*64 == NN*HID
}

// ---------------------------------------------------------------- input MLP
__global__ __launch_bounds__(32)
void k_inputnet(const float* __restrict__ x, const float* __restrict__ dn,
                const float* w1, const float* b1, const float* w2, const float* b2,
                const float* w3, const float* b3, float* __restrict__ out){
    int node = blockIdx.x, lane = threadIdx.x;
    __shared__ float l0[5], l1a[F2H], l2a[F2H];
    if (lane < 5) l0[lane] = x[node * 5 + lane] * dn[lane];
    __syncthreads();
    for (int o = lane; o < F2H; o += 32){
        float s = b1[o];
        for (int k = 0; k < 5; k++) s += l0[k] * w1[k * F2H + o];
        l1a[o] = eluf(s);
    }
    __syncthreads();
    for (int o = lane; o < F2H; o += 32){
        float s = b2[o];
        for (int k = 0; k < F2H; k++) s += l1a[k] * w2[k * F2H + o];
        l2a[o] = eluf(s);
    }
    __syncthreads();
    for (int o = lane; o < HID; o += 32){
        float s = b3[o];
        for (int k = 0; k < F2H; k++) s += l2a[k] * w3[k * HID + o];
        out[node * HID + o] = eluf(s);
    }
}

// ---------------------------------------------------------------- squared norms
__global__ void k_sq(const float* __restrict__ feat, float* __restrict__ sq){
    int i = blockIdx.x * blockDim.x + threadIdx.x;
    if (i < NN){
        const float* f = feat + (size_t)i * HID;
        float s = 0.f;
        for (int k = 0; k < HID; k++) s += f[k] * f[k];
        sq[i] = s;
    }
}

// ---------------------------------------------------------------- kNN (WMMA Gram + top-16)
__global__ __launch_bounds__(32)
void k_knn(const float* __restrict__ feat, const float* __restrict__ sq,
           const int* __restrict__ valid, int* __restrict__ nbr){
    __shared__ float At[16][HID];
    __shared__ float Bt[16][HID];
    __shared__ float rsq[16];
    __shared__ int   rval[16];
    __shared__ float d2s[16][16];
    __shared__ float topd[16][16];
    __shared__ int   topi[16][16];

    int lane     = threadIdx.x;
    int rowbase  = blockIdx.x * 16;
    int g        = rowbase / GN;
    int colstart = g * GN;

    for (int t = lane; t < 16 * HID; t += 32)
        At[t >> 6][t & 63] = feat[(size_t)(rowbase + (t >> 6)) * HID + (t & 63)];
    if (lane < 16){
        rsq[lane]  = sq[rowbase + lane];
        rval[lane] = valid[rowbase + lane];
        for (int k = 0; k < 16; k++){ topd[lane][k] = TOPINIT; topi[lane][k] = colstart; }
    }
    __syncthreads();

    unsigned ldsB = (unsigned)(uintptr_t)(&Bt[0][0]);   // low 32 bits = LDS offset
    int o  = (lane < 16) ? 0 : 2;
    int mb = (lane < 16) ? 0 : 8;

    for (int ct = 0; ct < GN / 16; ct++){
        int colbase = colstart + ct * 16;
        // async-stage the 16x64 f32 column tile into LDS (ASYNCcnt path)
        for (int it = 0; it < 8; ++it){
            int t = lane + it * 32;                    // 256 chunks of 16B
            const float* gp = feat + (size_t)(colbase + (t >> 4)) * HID + (t & 15) * 4;
            unsigned long long ga = (unsigned long long)(uintptr_t)gp;
            unsigned ldst = ldsB + (unsigned)t * 16u;
            asm volatile("global_load_async_to_lds_b128 %0, %1, off"
                         :: "v"(ldst), "v"(ga) : "memory");
        }
        asm volatile("s_wait_asynccnt 0" ::: "memory");
        __syncthreads();

        int   mycol = colbase + (lane & 15);
        float csq   = sq[mycol];
        int   cval  = valid[mycol];

        v8f c = {};
        for (int kc = 0; kc < 16; kc++){
            int kk = kc * 4 + o;
            v2f a; a.x = At[lane & 15][kk]; a.y = At[lane & 15][kk + 1];
            v2f b; b.x = Bt[lane & 15][kk]; b.y = Bt[lane & 15][kk + 1];
            c = wmma4(a, b, c);
        }
        for (int r = 0; r < 8; r++){
            int M = mb + r, row = rowbase + M;
            float d2 = rsq[M] - 2.0f * c[r] + csq;
            bool ok  = rval[M] && cval && (row != mycol);
            d2s[M][lane & 15] = ok ? d2 : BIGF;
        }
        __syncthreads();
        if (lane < 16){
            for (int nn2 = 0; nn2 < 16; nn2++){
                float d = d2s[lane][nn2];
                int idx = colbase + nn2;
                if (d < topd[lane][15]){
                    int p = 15;
                    while (p > 0 && d < topd[lane][p - 1]){
                        topd[lane][p] = topd[lane][p - 1];
                        topi[lane][p] = topi[lane][p - 1];
                        p--;
                    }
                    topd[lane][p] = d; topi[lane][p] = idx;
                }
            }
        }
        __syncthreads();
    }
    for (int t = lane; t < 16 * KNN; t += 32)
        nbr[(size_t)(rowbase + t / KNN) * KNN + (t % KNN)] = topi[t / KNN][t % KNN];
}

// ---------------------------------------------------------------- undirected adjacency (dedup via bitmask)
__global__ void k_adj(const int* __restrict__ nbr, const int* __restrict__ valid,
                      unsigned* __restrict__ adj){
    int i = blockIdx.x * blockDim.x + threadIdx.x;
    if (i >= NN || !valid[i]) return;
    int base = (i >> 11) << 11;
    int il = i - base;
    for (int k = 0; k < KNN; k++){
        int j = nbr[(size_t)i * KNN + k];
        if (j != i && valid[j]){
            int jl = j - base;
            atomicOr(&adj[(size_t)i * 64 + (jl >> 5)], 1u << (jl & 31));
            atomicOr(&adj[(size_t)j * 64 + (il >> 5)], 1u << (il & 31));
        }
    }
}

__global__ void k_deg(const unsigned* __restrict__ adj, float* __restrict__ invdeg){
    int i = blockIdx.x * blockDim.x + threadIdx.x;
    if (i >= NN) return;
    int cnt = 0;
    for (int w = 0; w < 64; w++) cnt += __popc(adj[(size_t)i * 64 + w]);
    invdeg[i] = (cnt > 0) ? 1.0f / (float)cnt : 0.0f;
}

// ---------------------------------------------------------------- EdgeConv (WMMA, wave per node)
__global__ __launch_bounds__(32)
void k_edgeconv(const float* __restrict__ feat, const unsigned* __restrict__ adj,
                const int* __restrict__ valid,
                const float* w1, const float* b1, const float* w2, const float* b2,
                float* __restrict__ xn){
    int i = blockIdx.x, lane = threadIdx.x;
    if (!valid[i]) return;                        // xn pre-cleared

    __shared__ int   nlist[GN];
    __shared__ int   jl[16];
    __shared__ int   em[16];
    __shared__ float xi[HID];
    __shared__ float A1[16][F2H];
    __shared__ float h1[16][F3H];
    __shared__ float red[2][16];
    __shared__ float xacc[HID];
    __shared__ int   degs;

    if (lane == 0){
        int base = (i >> 11) << 11, d = 0;
        for (int w = 0; w < 64; w++){
            unsigned m = adj[(size_t)i * 64 + w];
            while (m){ int b = __builtin_ctz(m); m &= m - 1; nlist[d++] = base + w * 32 + b; }
        }
        degs = d;
    }
    xi[lane]      = feat[(size_t)i * HID + lane];
    xi[lane + 32] = feat[(size_t)i * HID + lane + 32];
    xacc[lane] = 0.f; xacc[lane + 32] = 0.f;
    __syncthreads();
    int deg = degs;
    int ntile = (deg + 15) / 16;

    int col = lane & 15;
    int o   = (lane < 16) ? 0 : 2;
    int mb  = (lane < 16) ? 0 : 8;

    for (int t = 0; t < ntile; t++){
        if (lane < 16){
            int e = t * 16 + lane;
            int j = (e < deg) ? nlist[e] : -1;
            jl[lane] = j; em[lane] = (j >= 0);
            int e2 = e + 16;
            if (e2 < deg) __builtin_prefetch(&feat[(size_t)nlist[e2] * HID], 0, 1);
        }
        __syncthreads();
        for (int idx = lane; idx < 16 * HID; idx += 32){
            int m = idx >> 6, k = idx & 63;
            int j = jl[m];
            float xj = (j >= 0) ? feat[(size_t)j * HID + k] : 0.f;
            A1[m][k]       = xi[k];
            A1[m][HID + k] = xj - xi[k];
        }
        __syncthreads();

        // layer 1: [16 x 128] @ [128 x 96], ELU -> h1
        for (int nt = 0; nt < 6; nt++){
            int oc = nt * 16 + col;
            v8f c = {};
            for (int kc = 0; kc < 32; kc++){
                int kk = kc * 4 + o;
                v2f a; a.x = A1[col][kk];          a.y = A1[col][kk + 1];
                v2f b; b.x = w1[kk * F3H + oc];    b.y = w1[(kk + 1) * F3H + oc];
                c = wmma4(a, b, c);
            }
            float bias = b1[oc];
            for (int r = 0; r < 8; r++) h1[mb + r][oc] = eluf(c[r] + bias);
        }
        __syncthreads();

        // layer 2: [16 x 96] @ [96 x 64], ELU, edge-mask, row-sum accumulate
        for (int nt = 0; nt < 4; nt++){
            int oc = nt * 16 + col;
            v8f c = {};
            for (int kc = 0; kc < 24; kc++){
                int kk = kc * 4 + o;
                v2f a; a.x = h1[col][kk];          a.y = h1[col][kk + 1];
                v2f b; b.x = w2[kk * HID + oc];    b.y = w2[(kk + 1) * HID + oc];
                c = wmma4(a, b, c);
            }
            float bias = b2[oc], part = 0.f;
            for (int r = 0; r < 8; r++){
                float v = eluf(c[r] + bias);
                part += em[mb + r] ? v : 0.f;
            }
            red[lane >> 4][col] = part;
            __syncthreads();
            if (lane < 16) xacc[nt * 16 + lane] += red[0][lane] + red[1][lane];
            __syncthreads();
        }
    }
    xn[(size_t)i * HID + lane]      = xacc[lane];
    xn[(size_t)i * HID + lane + 32] = xacc[lane + 32];
}

// ---------------------------------------------------------------- graclus: best partner per node
__global__ __launch_bounds__(32)
void k_best(const float* __restrict__ xn, const unsigned* __restrict__ adj,
            const float* __restrict__ invdeg, const int* __restrict__ valid,
            int* __restrict__ partner){
    int i = blockIdx.x, lane = threadIdx.x;
    __shared__ float xi[HID];
    if (!valid[i]){ if (lane == 0) partner[i] = -1; return; }
    xi[lane]      = xn[(size_t)i * HID + lane];
    xi[lane + 32] = xn[(size_t)i * HID + lane + 32];
    __syncthreads();
    int base = (i >> 11) << 11;
    float inv_i = invdeg[i];

    float best = -BIGF;
    for (int w = lane; w < 64; w += 32){
        unsigned m = adj[(size_t)i * 64 + w];
        while (m){
            int b = __builtin_ctz(m); m &= m - 1;
            int j = base + w * 32 + b;
            float ss = 1e-12f;
            for (int k = 0; k < HID; k++){ float d = xi[k] - xn[(size_t)j * HID + k]; ss += d * d; }
            float we = sqrtf(ss) * (inv_i + invdeg[j]);
            best = fmaxf(best, we);
        }
    }
    for (int off = 16; off; off >>= 1) best = fmaxf(best, __shfl_xor(best, off, 32));

    int pm = 0x7FFFFFFF;
    for (int w = lane; w < 64; w += 32){
        unsigned m = adj[(size_t)i * 64 + w];
        while (m){
            int b = __builtin_ctz(m); m &= m - 1;
            int j = base + w * 32 + b;
            float ss = 1e-12f;
            for (int k = 0; k < HID; k++){ float d = xi[k] - xn[(size_t)j * HID + k]; ss += d * d; }
            float we = sqrtf(ss) * (inv_i + invdeg[j]);
            if (we >= best) pm = min(pm, j);
        }
    }
    for (int off = 16; off; off >>= 1) pm = min(pm, __shfl_xor(pm, off, 32));
    if (lane == 0) partner[i] = (pm != 0x7FFFFFFF) ? pm : -1;
}

__global__ void k_cluster(const int* __restrict__ partner, int* __restrict__ cluster){
    int i = blockIdx.x * blockDim.x + threadIdx.x;
    if (i >= NN) return;
    int p = partner[i];
    bool mutual = (p >= 0) && (partner[p] == i);
    cluster[i] = mutual ? min(i, p) : i;
}

// ---------------------------------------------------------------- max-pool into representative slots
__global__ void k_pool(const float* __restrict__ xn, const int* __restrict__ cluster,
                       const int* __restrict__ partner, const int* __restrict__ valid,
                       float* __restrict__ outf, int* __restrict__ outv){
    int s = blockIdx.x * blockDim.x + threadIdx.x;
    if (s >= NN) return;
    bool rep = (cluster[s] == s);
    int m1 = rep ? s : -1;
    int p  = partner[s];
    int m2 = (rep && p >= 0 && p != s && cluster[p] == s) ? p : -1;
    bool v1 = (m1 >= 0) && valid[m1];
    bool v2 = (m2 >= 0) && valid[m2];
    int nv = (v1 || v2) ? 1 : 0;
    for (int k = 0; k < HID; k++){
        float v = -BIGF;
        if (v1) v = fmaxf(v, xn[(size_t)m1 * HID + k]);
        if (v2) v = fmaxf(v, xn[(size_t)m2 * HID + k]);
        outf[(size_t)s * HID + k] = nv ? v : 0.f;
    }
    outv[s] = nv;
}

// ---------------------------------------------------------------- global max-pool + output MLP
__global__ void k_output(const float* __restrict__ x, const int* __restrict__ valid,
                         const float* w1, const float* b1, const float* w2, const float* b2,
                         const float* w3, const float* b3, float* __restrict__ out){
    int b = blockIdx.x, t = threadIdx.x;
    __shared__ float pm[4][HID];
    __shared__ float g[HID];
    __shared__ float o1[HID];
    __shared__ float o2[32];
    int k = t & 63, sg = t >> 6;
    float m = -BIGF;
    for (int s = b * GN + sg; s < (b + 1) * GN; s += 4)
        if (valid[s]) m = fmaxf(m, x[(size_t)s * HID + k]);
    pm[sg][k] = m;
    __syncthreads();
    if (t < HID) g[t] = fmaxf(fmaxf(pm[0][t], pm[1][t]), fmaxf(pm[2][t], pm[3][t]));
    __syncthreads();
    if (t < HID){
        float s = b1[t];
        for (int kk = 0; kk < HID; kk++) s += g[kk] * w1[kk * HID + t];
        o1[t] = eluf(s);
    }
    __syncthreads();
    if (t < 32){
        float s = b2[t];
        for (int kk = 0; kk < HID; kk++) s += o1[kk] * w2[kk * 32 + t];
        o2[t] = eluf(s);
    }
    __syncthreads();
    if (t == 0){
        float s = b3[0];
        for (int kk = 0; kk < 32; kk++) s += o2[kk] * w3[kk];
        out[b] = s;
    }
}

// ---------------------------------------------------------------- host driver
extern "C" void kernel_launch(void* const* d_in, const int* in_sizes, int n_in,
                              void* d_out, int out_size, void* d_ws, size_t ws_size,
                              hipStream_t stream){
    const float* x    = (const float*)d_in[0];
    // d_in[1] = batch (layout is fixed contiguous: 4 graphs x 2048)
    const float* dn   = (const float*)d_in[2];
    const float* iw1  = (const float*)d_in[3];  const float* ib1 = (const float*)d_in[4];
    const float* iw2  = (const float*)d_in[5];  const float* ib2 = (const float*)d_in[6];
    const float* iw3  = (const float*)d_in[7];  const float* ib3 = (const float*)d_in[8];
    const float* c1w1 = (const float*)d_in[9];  const float* c1b1 = (const float*)d_in[10];
    const float* c1w2 = (const float*)d_in[11]; const float* c1b2 = (const float*)d_in[12];
    const float* c2w1 = (const float*)d_in[13]; const float* c2b1 = (const float*)d_in[14];
    const float* c2w2 = (const float*)d_in[15]; const float* c2b2 = (const float*)d_in[16];
    const float* ow1  = (const float*)d_in[17]; const float* ob1 = (const float*)d_in[18];
    const float* ow2  = (const float*)d_in[19]; const float* ob2 = (const float*)d_in[20];
    const float* ow3  = (const float*)d_in[21]; const float* ob3 = (const float*)d_in[22];

    char* w = (char*)d_ws;
    float*    feat0  = (float*)w;     w += (size_t)NN * HID * 4;
    float*    featxn = (float*)w;     w += (size_t)NN * HID * 4;
    float*    featnx = (float*)w;     w += (size_t)NN * HID * 4;
    float*    sq     = (float*)w;     w += (size_t)NN * 4;
    int*      validA = (int*)w;       w += (size_t)NN * 4;
    int*      validB = (int*)w;       w += (size_t)NN * 4;
    int*      nbr    = (int*)w;       w += (size_t)NN * KNN * 4;
    unsigned* adj    = (unsigned*)w;  w += (size_t)NN * 64 * 4;
    float*    invdeg = (float*)w;     w += (size_t)NN * 4;
    int*      partner= (int*)w;       w += (size_t)NN * 4;
    int*      cluster= (int*)w;       /* end */

    k_init_valid<<<NN / 256, 256, 0, stream>>>(validA);
    k_inputnet<<<NN, 32, 0, stream>>>(x, dn, iw1, ib1, iw2, ib2, iw3, ib3, feat0);

    float* fc = feat0;  float* fn = featnx;
    int*   vc = validA; int*   vn = validB;
    for (int r = 0; r < 3; r++){
        const float* cw1 = (r < 2) ? c1w1 : c2w1;
        const float* cb1 = (r < 2) ? c1b1 : c2b1;
        const float* cw2 = (r < 2) ? c1w2 : c2w2;
        const float* cb2 = (r < 2) ? c1b2 : c2b2;

        k_clear<<<(NN * 64) / 256, 256, 0, stream>>>(adj, featxn);
        k_sq<<<NN / 256, 256, 0, stream>>>(fc, sq);
        k_knn<<<NN / 16, 32, 0, stream>>>(fc, sq, vc, nbr);
        k_adj<<<NN / 256, 256, 0, stream>>>(nbr, vc, adj);
        k_deg<<<NN / 256, 256, 0, stream>>>(adj, invdeg);
        k_edgeconv<<<NN, 32, 0, stream>>>(fc, adj, vc, cw1, cb1, cw2, cb2, featxn);
        k_best<<<NN, 32, 0, stream>>>(featxn, adj, invdeg, vc, partner);
        k_cluster<<<NN / 256, 256, 0, stream>>>(partner, cluster);
        k_pool<<<NN / 256, 256, 0, stream>>>(featxn, cluster, partner, vc, fn, vn);

        float* tf = fc; fc = fn; fn = tf;
        int*   tv = vc; vc = vn; vn = tv;
    }
    k_output<<<4, 256, 0, stream>>>(fc, vc, ow1, ob1, ow2, ob2, ow3, ob3, (float*)d_out);
}